// Attention_23244363005972
// MI455X (gfx1250) — compile-verified
//
#include <hip/hip_runtime.h>
#include <hip/hip_bf16.h>

#define SEQ_LEN 2048
#define BATCH   64
#define HIDDEN  1024

typedef __attribute__((ext_vector_type(2))) float v2f;
typedef __attribute__((ext_vector_type(8))) float v8f;

// ---------------------------------------------------------------------------
// Stage 1: hq[64,1024] = hidden[64,1024] @ W[1024,1024]   (full f32 via WMMA)
// One wave (32 threads) per 16x16 output tile. K marched in steps of 8 using
// two independent v_wmma_f32_16x16x4_f32 accumulator chains for ILP.
// A-frag layout (ISA 7.12.2, 32-bit A 16x4): lanes 0-15 -> K=k,k+1 for row M=l;
// lanes 16-31 -> K=k+2,k+3. B-frag mirrors it (rows of W striped across lanes).
// C/D layout: VGPR r -> row m0+r (lanes 0-15) / m0+8+r (lanes 16-31), col n0+l.
// ---------------------------------------------------------------------------
__global__ void hq_gemm_wmma_f32(const float* __restrict__ hidden,
                                 const float* __restrict__ W,
                                 float* __restrict__ hq) {
    const int lane = threadIdx.x;      // 0..31, full wave, EXEC all ones
    const int half = lane >> 4;        // 0 or 1
    const int l    = lane & 15;
    const int n0   = blockIdx.x * 16;  // 64 tiles over N
    const int m0   = blockIdx.y * 16;  // 4 tiles over M

    v8f c0 = {0.f, 0.f, 0.f, 0.f, 0.f, 0.f, 0.f, 0.f};
    v8f c1 = {0.f, 0.f, 0.f, 0.f, 0.f, 0.f, 0.f, 0.f};

    const float* __restrict__ arow = hidden + (size_t)(m0 + l) * HIDDEN;

    for (int k = 0; k < HIDDEN; k += 8) {
        const int ka = k + 2 * half;       // K pair for accumulator 0
        const int kb = k + 4 + 2 * half;   // K pair for accumulator 1

        v2f a0, b0, a1, b1;
        a0.x = arow[ka];
        a0.y = arow[ka + 1];
        b0.x = W[(size_t)ka       * HIDDEN + n0 + l];
        b0.y = W[(size_t)(ka + 1) * HIDDEN + n0 + l];

        a1.x = arow[kb];
        a1.y = arow[kb + 1];
        b1.x = W[(size_t)kb       * HIDDEN + n0 + l];
        b1.y = W[(size_t)(kb + 1) * HIDDEN + n0 + l];

        // 8 args: (neg_a, A, neg_b, B, c_mod, C, reuse_a, reuse_b)
        c0 = __builtin_amdgcn_wmma_f32_16x16x4_f32(false, a0, false, b0,
                                                   (short)0, c0, false, false);
        c1 = __builtin_amdgcn_wmma_f32_16x16x4_f32(false, a1, false, b1,
                                                   (short)0, c1, false, false);
    }

    v8f c = c0 + c1;
    const int row_base = m0 + 8 * half;
#pragma unroll
    for (int r = 0; r < 8; ++r) {
        hq[(size_t)(row_base + r) * HIDDEN + n0 + l] = c[r];
    }
}

// ---------------------------------------------------------------------------
// Stage 2: energies[i,b] = dot(hq[b,:], enc[i,b,:])   -- the 512 MB stream.
// Block = 256 threads (8 waves), blockIdx.y = batch b, blockIdx.x = seq chunk
// of 8 rows (one wave per row). hq[b] (4 KB) cached in LDS once per block.
// Each wave issues coalesced float4 (b128) loads: 8 iters x 128 B/lane-group.
// ---------------------------------------------------------------------------
__global__ void energies_dot(const float* __restrict__ enc,
                             const float* __restrict__ hq,
                             float* __restrict__ energies) {
    __shared__ float4 s_hq[HIDDEN / 4];   // 256 float4 = 4 KB

    const int b   = blockIdx.y;
    const int tid = threadIdx.x;          // 0..255

    const float4* __restrict__ hqv =
        reinterpret_cast<const float4*>(hq + (size_t)b * HIDDEN);
    s_hq[tid] = hqv[tid];
    __syncthreads();

    const int wave = tid >> 5;
    const int lane = tid & 31;
    const int i    = blockIdx.x * 8 + wave;

    const float4* __restrict__ encv = reinterpret_cast<const float4*>(
        enc + ((size_t)i * BATCH + b) * HIDDEN);

    float sum = 0.f;
#pragma unroll
    for (int it = 0; it < 8; ++it) {
        const int idx = it * 32 + lane;
        const float4 e = encv[idx];
        const float4 h = s_hq[idx];
        sum += e.x * h.x + e.y * h.y + e.z * h.z + e.w * h.w;
    }

    // wave32 reduction
#pragma unroll
    for (int off = 16; off > 0; off >>= 1)
        sum += __shfl_down(sum, off, 32);

    if (lane == 0)
        energies[(size_t)i * BATCH + b] = sum;
}

// ---------------------------------------------------------------------------
// Stage 3: softmax over axis 0 (seq) per batch column, in-place on d_out.
// One block per batch column b; 256 threads x 8 seq rows each. All 8 values
// are read into registers before any write -> in-place is race-free.
// ---------------------------------------------------------------------------
__global__ void softmax_seq(float* __restrict__ e) {
    __shared__ float red_max[8];
    __shared__ float red_sum[8];

    const int b    = blockIdx.x;
    const int tid  = threadIdx.x;        // 0..255
    const int wave = tid >> 5;
    const int lane = tid & 31;

    float vals[8];
    float m = -__FLT_MAX__;
#pragma unroll
    for (int j = 0; j < 8; ++j) {
        vals[j] = e[(size_t)(tid + j * 256) * BATCH + b];
        m = fmaxf(m, vals[j]);
    }
#pragma unroll
    for (int off = 16; off > 0; off >>= 1)
        m = fmaxf(m, __shfl_down(m, off, 32));
    if (lane == 0) red_max[wave] = m;
    __syncthreads();

    float gmax = red_max[0];
#pragma unroll
    for (int w = 1; w < 8; ++w) gmax = fmaxf(gmax, red_max[w]);

    float s = 0.f;
#pragma unroll
    for (int j = 0; j < 8; ++j) {
        vals[j] = __expf(vals[j] - gmax);
        s += vals[j];
    }
#pragma unroll
    for (int off = 16; off > 0; off >>= 1)
        s += __shfl_down(s, off, 32);
    if (lane == 0) red_sum[wave] = s;
    __syncthreads();

    float gsum = red_sum[0];
#pragma unroll
    for (int w = 1; w < 8; ++w) gsum += red_sum[w];
    const float inv = 1.0f / gsum;

#pragma unroll
    for (int j = 0; j < 8; ++j)
        e[(size_t)(tid + j * 256) * BATCH + b] = vals[j] * inv;
}

extern "C" void kernel_launch(void* const* d_in, const int* in_sizes, int n_in,
                              void* d_out, int out_size, void* d_ws, size_t ws_size,
                              hipStream_t stream) {
    const float* hidden = (const float*)d_in[0];   // [1, 64, 1024]
    const float* enc    = (const float*)d_in[1];   // [2048, 64, 1024]
    const float* W      = (const float*)d_in[2];   // [1024, 1024]
    float* out = (float*)d_out;                    // [2048, 64] energies -> softmax
    float* hq  = (float*)d_ws;                     // [64, 1024] scratch (256 KB)

    // Stage 1: hq = hidden @ W   (f32 WMMA, exact)
    hq_gemm_wmma_f32<<<dim3(HIDDEN / 16, 64 / 16), 32, 0, stream>>>(hidden, W, hq);

    // Stage 2: energies[i,b] = <hq[b], enc[i,b]>  (bandwidth-bound stream)
    energies_dot<<<dim3(SEQ_LEN / 8, BATCH), 256, 0, stream>>>(enc, hq, out);

    // Stage 3: softmax over seq axis, in-place on d_out
    softmax_seq<<<BATCH, 256, 0, stream>>>(out);
}